// RotationParamModel_29996051595547
// MI455X (gfx1250) — compile-verified
//
#include <hip/hip_runtime.h>
#include <hip/hip_bf16.h>
#include <math.h>

typedef __attribute__((ext_vector_type(2))) float v2f;
typedef __attribute__((ext_vector_type(8))) float v8f;

#define NSAMP 8192
#define DD    32
#define HH    512
#define NPAR  30

// LDS row strides (floats): stride % 32 = 4 avoids bank conflicts for the
// 16-row fragment loads, and even stride keeps ds_load_b64 8B-aligned.
#define XS 36
#define HS 516
#define PS 36

__device__ __forceinline__ v8f wmma4(v2f a, v2f b, v8f c) {
    return __builtin_amdgcn_wmma_f32_16x16x4_f32(false, a, false, b, (short)0, c, false, false);
}

// ---------------------------------------------------------------------------
// Householder QR of 32x32 A (LAPACK sign convention), single thread (tiny).
// ---------------------------------------------------------------------------
__global__ void qr_kernel(const float* __restrict__ A, float* __restrict__ Q) {
    if (threadIdx.x != 0 || blockIdx.x != 0) return;
    float R[DD][DD];
    float Qm[DD][DD];
    for (int i = 0; i < DD; ++i)
        for (int j = 0; j < DD; ++j) {
            R[i][j] = A[i * DD + j];
            Qm[i][j] = (i == j) ? 1.f : 0.f;
        }
    for (int k = 0; k < DD; ++k) {
        float nrm2 = 0.f;
        for (int i = k; i < DD; ++i) nrm2 += R[i][k] * R[i][k];
        float nrm = sqrtf(nrm2);
        if (nrm == 0.f) continue;
        float alpha = (R[k][k] >= 0.f) ? -nrm : nrm;   // LAPACK convention
        float v[DD];
        for (int i = 0; i < DD; ++i) v[i] = (i < k) ? 0.f : R[i][k];
        v[k] -= alpha;
        float vn2 = 0.f;
        for (int i = k; i < DD; ++i) vn2 += v[i] * v[i];
        if (vn2 < 1e-30f) continue;
        float inv2 = 2.f / vn2;
        for (int j = k; j < DD; ++j) {
            float dot = 0.f;
            for (int i = k; i < DD; ++i) dot += v[i] * R[i][j];
            dot *= inv2;
            for (int i = k; i < DD; ++i) R[i][j] -= v[i] * dot;
        }
        for (int i = 0; i < DD; ++i) {
            float dot = 0.f;
            for (int j = k; j < DD; ++j) dot += Qm[i][j] * v[j];
            dot *= inv2;
            for (int j = k; j < DD; ++j) Qm[i][j] -= dot * v[j];
        }
    }
    for (int i = 0; i < DD; ++i)
        for (int j = 0; j < DD; ++j)
            Q[i * DD + j] = Qm[i][j];
}

// ---------------------------------------------------------------------------
// Pair-interleaved transpose: logical B[k][n] = src[n][k] (n<R, zero-pad to Rp),
// stored so that (B[k][n], B[k+1][n]) for even k are adjacent:
//   dst[(k>>1)*(Rp*2) + n*2 + (k&1)]
// -> one coalesced global_load_b64 per lane per WMMA k-step.
// ---------------------------------------------------------------------------
__global__ void transpose_pair(const float* __restrict__ src, float* __restrict__ dst,
                               int R, int C, int Rp) {
    int idx = blockIdx.x * blockDim.x + threadIdx.x;
    if (idx >= C * Rp) return;
    int c = idx / Rp;            // k index (0..C-1)
    int r = idx - c * Rp;        // n index (0..Rp-1)
    float v = (r < R) ? src[r * C + c] : 0.f;
    dst[(c >> 1) * (Rp * 2) + r * 2 + (c & 1)] = v;
}

// ---------------------------------------------------------------------------
// xp = x @ Q   (8192x32 @ 32x32)
// ---------------------------------------------------------------------------
__global__ void rotate_kernel(const float* __restrict__ x, const float* __restrict__ Q,
                              float* __restrict__ xp) {
    int idx = blockIdx.x * blockDim.x + threadIdx.x;
    if (idx >= NSAMP * DD) return;
    int n = idx >> 5, j = idx & 31;
    float s = 0.f;
#pragma unroll
    for (int d = 0; d < DD; ++d) s += x[n * DD + d] * Q[d * DD + j];
    xp[idx] = s;
}

// ---------------------------------------------------------------------------
// One 32xH layer on WMMA f32 16x16x4:  out = act(32xK) @ W(KxH) + bias.
// Each wave owns 4 consecutive N-tiles (8 independent accumulator chains);
// A-fragments (LDS, ds_load_b64) are reused across all 4 N-tiles; B-fragments
// come from the pair-interleaved weight array with one global_load_b64 each.
// Fragment layout (ISA 7.12.2, 32-bit):
//   A: lane L -> row M = L&15 ; VGPR j -> K = k0 + ((L>>4)<<1) + j
//   B: lane L -> col N = L&15 ; same K mapping
//   C: VGPR r -> M = r + 8*(L>>4), N = L&15
// ---------------------------------------------------------------------------
__device__ __forceinline__ void gemm_layer(const float* __restrict__ wtp,
                                           const float* __restrict__ bias,
                                           const float* act, int actS, int Kdim,
                                           float* out, int outS,
                                           bool relu, int wave, int ln) {
    const int mrow  = ln & 15;
    const int klane = (ln >> 4) << 1;
    const int mhi   = (ln >> 4) << 3;
    const int col0  = wave * 64 + mrow;   // first of 4 N-tiles for this wave
    v8f c0[4] = {};
    v8f c1[4] = {};
#pragma unroll 2
    for (int k0 = 0; k0 < Kdim; k0 += 4) {
        const int ka = k0 + klane;
        float2 fa0 = *(const float2*)&act[mrow * actS + ka];
        float2 fa1 = *(const float2*)&act[(16 + mrow) * actS + ka];
        v2f a0; a0[0] = fa0.x; a0[1] = fa0.y;
        v2f a1; a1[0] = fa1.x; a1[1] = fa1.y;
        const float* wrow = &wtp[(ka >> 1) * (2 * HH) + col0 * 2];
#pragma unroll
        for (int q = 0; q < 4; ++q) {
            float2 fb = *(const float2*)(wrow + 32 * q);
            v2f b; b[0] = fb.x; b[1] = fb.y;
            c0[q] = wmma4(a0, b, c0[q]);
            c1[q] = wmma4(a1, b, c1[q]);
        }
    }
#pragma unroll
    for (int q = 0; q < 4; ++q) {
        const int col = col0 + 16 * q;
        const float bv = bias[col];
#pragma unroll
        for (int r = 0; r < 8; ++r) {
            float v0 = c0[q][r] + bv;
            float v1 = c1[q][r] + bv;
            if (relu) { v0 = fmaxf(v0, 0.f); v1 = fmaxf(v1, 0.f); }
            out[(mhi + r) * outS + col]      = v0;
            out[(16 + mhi + r) * outS + col] = v1;
        }
    }
}

// ---------------------------------------------------------------------------
// Fused per-sample: mask -> L1 -> L2 -> L3 -> logistic-mixture loglik partial.
// One block per sample, 256 threads (8 wave32).
// ---------------------------------------------------------------------------
__global__ __launch_bounds__(256, 1)
void fused_kernel(const float* __restrict__ xp,
                  const float* __restrict__ wt1, const float* __restrict__ b1,
                  const float* __restrict__ wt2, const float* __restrict__ b2,
                  const float* __restrict__ wt3, const float* __restrict__ b3,
                  float* __restrict__ partials) {
    __shared__ float xm[32 * XS];
    __shared__ float h1[32 * HS];
    __shared__ float h2[32 * HS];
    __shared__ float ppA[32 * PS];
    __shared__ float ppB[32 * PS];
    __shared__ float xps[32];
    __shared__ float b3p[32];

    const int t = threadIdx.x;
    const int wave = t >> 5;
    const int ln = t & 31;
    const int n = blockIdx.x;

    // Stage masked activation block: xm[i][j] = xp[n][j] * (j < i)
    for (int idx = t; idx < 32 * 32; idx += 256) {
        int i = idx >> 5, j = idx & 31;
        xm[i * XS + j] = (j < i) ? xp[n * DD + j] : 0.f;
    }
    if (t < 32) {
        xps[t] = xp[n * DD + t];
        b3p[t] = (t < NPAR) ? b3[t] : 0.f;
    }
    __syncthreads();

    // Layer 1: h1 = relu(xm @ W1^T + b1), K = 32
    gemm_layer(wt1, b1, xm, XS, DD, h1, HS, true, wave, ln);
    __syncthreads();

    // Layer 2: h2 = relu(h1 @ W2^T + b2), K = 512
    gemm_layer(wt2, b2, h1, HS, HH, h2, HS, true, wave, ln);
    __syncthreads();

    // Layer 3: pp(32x32) = h2 @ W3^T + b3 (N padded to 32).
    // 4 output tiles x 2 K-halves -> all 8 waves busy; halves summed later.
    {
        const int tile  = wave & 3;
        const int mt    = tile >> 1, nt = tile & 1;
        const int khalf = wave >> 2;
        const int m0 = mt * 16, n0 = nt * 16;
        const int mrow = ln & 15;
        const int col  = n0 + mrow;
        const int klane = (ln >> 4) << 1;
        v8f c = {};
        const int kbeg = khalf * (HH / 2);
#pragma unroll 2
        for (int k0 = kbeg; k0 < kbeg + HH / 2; k0 += 4) {
            const int ka = k0 + klane;
            float2 fa = *(const float2*)&h2[(m0 + mrow) * HS + ka];
            v2f a; a[0] = fa.x; a[1] = fa.y;
            float2 fb = *(const float2*)&wt3[(ka >> 1) * 64 + col * 2];
            v2f b; b[0] = fb.x; b[1] = fb.y;
            c = wmma4(a, b, c);
        }
        const int mhi = (ln >> 4) << 3;
        float* dst = khalf ? ppB : ppA;
        const float bv = khalf ? 0.f : b3p[col];
#pragma unroll
        for (int r = 0; r < 8; ++r)
            dst[(m0 + mhi + r) * PS + col] = c[r] + bv;
    }
    __syncthreads();

    // Mixture-of-logistics loglik: one lane per row i, wave 0 only.
    if (t < 32) {
        float P[NPAR];
#pragma unroll
        for (int k = 0; k < NPAR; ++k)
            P[k] = ppA[t * PS + k] + ppB[t * PS + k];
        const float xv = xps[t];
        float mx = P[0];
#pragma unroll
        for (int k = 1; k < 10; ++k) mx = fmaxf(mx, P[k]);
        float we[10];
        float se = 0.f;
#pragma unroll
        for (int k = 0; k < 10; ++k) { we[k] = expf(P[k] - mx); se += we[k]; }
        const float inv_se = 1.f / se;
        float acc = 0.f;
#pragma unroll
        for (int k = 0; k < 10; ++k) {
            float ls = fminf(fmaxf(P[20 + k], -7.f), 7.f);
            float is = expf(-ls);
            float z = (xv - P[10 + k]) * is;
            float sg = 1.f / (1.f + expf(-z));
            acc += (we[k] * inv_se) * sg * (1.f - sg) * is;
        }
        float ll = logf(acc + 1e-10f);
#pragma unroll
        for (int off = 16; off > 0; off >>= 1) ll += __shfl_xor(ll, off);
        if (ln == 0) partials[n] = ll;
    }
}

// ---------------------------------------------------------------------------
// Deterministic fixed-order final reduction.
// ---------------------------------------------------------------------------
__global__ void reduce_kernel(const float* __restrict__ partials, int n,
                              float* __restrict__ out) {
    __shared__ float s[256];
    float acc = 0.f;
    for (int i = threadIdx.x; i < n; i += 256) acc += partials[i];
    s[threadIdx.x] = acc;
    __syncthreads();
    for (int off = 128; off > 0; off >>= 1) {
        if ((int)threadIdx.x < off) s[threadIdx.x] += s[threadIdx.x + off];
        __syncthreads();
    }
    if (threadIdx.x == 0) out[0] = s[0];
}

extern "C" void kernel_launch(void* const* d_in, const int* in_sizes, int n_in,
                              void* d_out, int out_size, void* d_ws, size_t ws_size,
                              hipStream_t stream) {
    const float* x  = (const float*)d_in[0];
    const float* A  = (const float*)d_in[1];
    const float* W1 = (const float*)d_in[2];
    const float* b1 = (const float*)d_in[3];
    const float* W2 = (const float*)d_in[4];
    const float* b2 = (const float*)d_in[5];
    const float* W3 = (const float*)d_in[6];
    const float* b3 = (const float*)d_in[7];
    float* out = (float*)d_out;

    float* Q        = (float*)d_ws;            // 32*32
    float* xp       = Q + 1024;                // 8192*32
    float* wt1      = xp + NSAMP * DD;         // 32*512   (pair-interleaved)
    float* wt2      = wt1 + DD * HH;           // 512*512  (pair-interleaved)
    float* wt3      = wt2 + HH * HH;           // 512*32   (pair-interleaved, N padded)
    float* partials = wt3 + HH * 32;           // 8192

    qr_kernel<<<1, 1, 0, stream>>>(A, Q);
    transpose_pair<<<(DD * HH + 255) / 256, 256, 0, stream>>>(W1, wt1, HH, DD, HH);
    transpose_pair<<<(HH * HH + 255) / 256, 256, 0, stream>>>(W2, wt2, HH, HH, HH);
    transpose_pair<<<(HH * 32 + 255) / 256, 256, 0, stream>>>(W3, wt3, NPAR, HH, 32);
    rotate_kernel<<<(NSAMP * DD + 255) / 256, 256, 0, stream>>>(x, Q, xp);
    fused_kernel<<<NSAMP, 256, 0, stream>>>(xp, wt1, b1, wt2, b2, wt3, b3, partials);
    reduce_kernel<<<1, 256, 0, stream>>>(partials, NSAMP, out);
}